// SkipConnection_89412629168745
// MI455X (gfx1250) — compile-verified
//
#include <hip/hip_runtime.h>

// CDNA5 / gfx1250: wave32, WMMA f32 16x16x4.
typedef __attribute__((ext_vector_type(2))) float v2f;  // A / B operands (2 VGPRs)
typedef __attribute__((ext_vector_type(8))) float v8f;  // C / D operands (8 VGPRs)

#define N_LAYERS 99
#define TPW 4            // independent 16-row tiles per wave (hides WMMA latency)

// Single-instruction ReLU: inputs are finite, so skip fmaxf's NaN-canonicalize.
__device__ __forceinline__ float relu1(float x) {
    float r;
    asm("v_max_num_f32 %0, 0, %1" : "=v"(r) : "v"(x));
    return r;
}

// One 4x4 layer applied to all TPW 16-row tiles via V_WMMA_F32_16X16X4_F32.
//
// A carries W^T twice — rows 0..3 = W^T rows 0..3, row 8 = W^T row 2, row 9 =
// W^T row 3. Since D VGPR j holds M=j in lanes 0..15 and M=j+8 in lanes 16..31,
// d[0] = {feat0 | feat2} and d[1] = {feat1 | feat3} — exactly the next layer's
// B operand layout: the spare WMMA rows do the cross-lane repack for free.
//
// C is a zero splat so it folds to the inline SRC2=0 encoding (no C registers,
// no zero-building movs, WMMA depends only on A/B). Bias is added post-WMMA
// with the lane-varying pair {b0|b2},{b1|b3} from the LDS entry.
//
// `e` holds this layer's {a0,a1,cb0,cb1}; the step prefetches the next layer's
// entry (one full layer of slack to cover LDS latency).
//
// MODE: 0 = no residual, 1 = out += carried skip, 2 = out += out,
//       3 = out += out and save skip (last layer of a residual group).
template <int MODE>
__device__ __forceinline__ void layer_step(
    int inext, const float* __restrict__ ldsT, int lane, float4& e,
    float (&h0)[TPW], float (&h1)[TPW],
    float (&p0)[TPW], float (&p1)[TPW])
{
    const float4 cur = e;
    e = *(const float4*)&ldsT[inext * 128 + lane * 4];   // prefetch next layer

    v2f a; a.x = cur.x; a.y = cur.y;
    const float cb0 = cur.z;                 // bias: {b0|b2}
    const float cb1 = cur.w;                 // bias: {b1|b3}
    const v8f cz = {0.0f, 0.0f, 0.0f, 0.0f, 0.0f, 0.0f, 0.0f, 0.0f};

    #pragma unroll
    for (int t = 0; t < TPW; ++t) {
        v2f bm; bm.x = h0[t]; bm.y = h1[t];
        v8f d = __builtin_amdgcn_wmma_f32_16x16x4_f32(
            false, a, false, bm, (short)0, cz, false, false);
        float n0 = relu1(d[0] + cb0);        // D already in next-B layout
        float n1 = relu1(d[1] + cb1);
        if (MODE == 1) { n0 += p0[t]; n1 += p1[t]; }
        if (MODE == 2 || MODE == 3) { n0 += h0[t]; n1 += h1[t]; }
        if (MODE == 3) { p0[t] = n0; p1[t] = n1; }
        h0[t] = n0; h1[t] = n1;
    }
}

__global__ __launch_bounds__(256) void skipmlp_wmma_kernel(
    const float* __restrict__ x,
    const float* __restrict__ W_in,
    const float* __restrict__ b_in,
    const float* __restrict__ Ws,
    const float* __restrict__ bs,
    const float* __restrict__ W_out,
    const float* __restrict__ b_out,
    float* __restrict__ out,
    int nrows)
{
    // Per-layer lane table: {a0, a1, cb0, cb1} x 32 lanes x 99 layers (+1 dummy
    // row so the last step can prefetch harmlessly). ~50 KB of 320 KB LDS.
    __shared__ float ldsT[(N_LAYERS + 1) * 128];

    const int tid = threadIdx.x;
    for (int idx = tid; idx < N_LAYERS * 32; idx += 256) {
        const int L  = idx >> 5;
        const int ln = idx & 31;
        const int mm = ln & 15;
        const int hh = ln >> 4;
        const int k0 = 2 * hh;
        int r = -1;                           // duplicated-W^T row map
        if (mm < 4)       r = mm;
        else if (mm == 8) r = 2;
        else if (mm == 9) r = 3;
        float a0 = 0.0f, a1 = 0.0f;
        if (r >= 0) {
            a0 = Ws[L * 16 + (k0 + 0) * 4 + r];   // W^T[row, k] = W[k, row]
            a1 = Ws[L * 16 + (k0 + 1) * 4 + r];
        }
        const float c0 = bs[L * 4 + k0 + 0];      // bias for this lane half
        const float c1 = bs[L * 4 + k0 + 1];
        ldsT[L * 128 + ln * 4 + 0] = a0;
        ldsT[L * 128 + ln * 4 + 1] = a1;
        ldsT[L * 128 + ln * 4 + 2] = c0;
        ldsT[L * 128 + ln * 4 + 3] = c1;
    }
    __syncthreads();

    const int lane = tid & 31;
    const int m    = lane & 15;          // row-within-tile (N index)
    const bool lowHalf = (lane < 16);    // low half: features {0,1}; high: {2,3}
    const int k0 = lowHalf ? 0 : 2;
    const int k1 = k0 + 1;

    // Per-lane input/output layer constants (loaded once).
    const float wi00 = W_in[0 * 4 + k0];
    const float wi10 = W_in[1 * 4 + k0];
    const float wi01 = W_in[0 * 4 + k1];
    const float wi11 = W_in[1 * 4 + k1];
    const float bi0  = b_in[k0];
    const float bi1  = b_in[k1];
    const float wo0  = W_out[k0];
    const float wo1  = W_out[k1];
    const float bo   = b_out[0];

    const int waveId = blockIdx.x * (blockDim.x >> 5) + (tid >> 5);
    const long long rowBase = (long long)waveId * (TPW * 16);

    // Activations in B-matrix (4x16) layout: h0 = V0 = {K0|K2}, h1 = V1 = {K1|K3}.
    // p0/p1 carry the skip across each group's non-residual layer.
    float h0[TPW], h1[TPW], p0[TPW], p1[TPW];

    #pragma unroll
    for (int t = 0; t < TPW; ++t) {
        long long row  = rowBase + t * 16 + m;
        long long lrow = (row < nrows) ? row : (long long)(nrows - 1);  // clamp: EXEC stays full
        const v2f xv = __builtin_nontemporal_load((const v2f*)(x + lrow * 2));
        float v0 = relu1(xv.x * wi00 + xv.y * wi10 + bi0);
        float v1 = relu1(xv.x * wi01 + xv.y * wi11 + bi1);
        h0[t] = v0; h1[t] = v1;
        p0[t] = v0; p1[t] = v1;   // skip = relu(x@W_in + b_in)
    }

    // Static residual schedule: layer 5g has no residual; 5g+1 adds the skip
    // carried from the previous group; 5g+2..5g+4 have skip == out. The last
    // layer of each full group saves the skip for the next group (MODE 3).
    float4 e = *(const float4*)&ldsT[lane * 4];   // layer 0 entry
    for (int g = 0; g < 20; ++g) {
        const int i = 5 * g;
        layer_step<0>(i + 1, ldsT, lane, e, h0, h1, p0, p1);   // layer i
        layer_step<1>(i + 2, ldsT, lane, e, h0, h1, p0, p1);   // layer i+1
        layer_step<2>(i + 3, ldsT, lane, e, h0, h1, p0, p1);   // layer i+2
        layer_step<2>(i + 4, ldsT, lane, e, h0, h1, p0, p1);   // layer i+3
        if (i + 4 < N_LAYERS)                                  // layer i+4
            layer_step<3>(i + 5, ldsT, lane, e, h0, h1, p0, p1);
    }

    // out @ W_out + b_out : per-lane partial, cross-half reduce, lanes 0..15 store.
    #pragma unroll
    for (int t = 0; t < TPW; ++t) {
        float p = h0[t] * wo0 + h1[t] * wo1;
        float q = __shfl(p, m + 16, 32);        // lanes 0..15 grab upper-half partner
        float y = p + q + bo;
        long long row = rowBase + t * 16 + m;
        if (lowHalf && row < nrows)
            __builtin_nontemporal_store(y, out + row);
    }
}

extern "C" void kernel_launch(void* const* d_in, const int* in_sizes, int n_in,
                              void* d_out, int out_size, void* d_ws, size_t ws_size,
                              hipStream_t stream) {
    const float* x     = (const float*)d_in[0];
    const float* W_in  = (const float*)d_in[1];
    const float* b_in  = (const float*)d_in[2];
    const float* Ws    = (const float*)d_in[3];
    const float* bs    = (const float*)d_in[4];
    const float* W_out = (const float*)d_in[5];
    const float* b_out = (const float*)d_in[6];
    float* out = (float*)d_out;

    const int nrows = in_sizes[0] / 2;                 // x is [B,2]
    const int rowsPerWave = TPW * 16;
    const int waves  = (nrows + rowsPerWave - 1) / rowsPerWave;
    const int blocks = (waves + 7) / 8;                // 256 threads = 8 wave32s

    skipmlp_wmma_kernel<<<blocks, 256, 0, stream>>>(
        x, W_in, b_in, Ws, bs, W_out, b_out, out, nrows);
}